// Cluster_loss_67276367725273
// MI455X (gfx1250) — compile-verified
//
#include <hip/hip_runtime.h>

// ---------------------------------------------------------------------------
// Cluster (discriminative) loss for MI455X / gfx1250.
//   features: (N=4, C=32, H=256, W=512) fp32, ground_truth: (4, 256*512) int
//   outputs:  [total, variance, distance, normalization, mu(4x32x16)] fp32
//
// ws layout (floats): [0,2048) mu_sum(n,c,k) ; [2048,2112) S1(n,k) ; [2112,2176) cnt(n,k)
// ---------------------------------------------------------------------------

#define N_B   4
#define C_CH  32
#define K_CL  16
#define P_PIX (256 * 512)

#define BLOCKS_PER_N 16   // pass-1 blocks per image (8 waves each)
#define B2           64   // pass-2 blocks per image

__constant__ const float kDeltaV = 0.2f;
__constant__ const float kDeltaD = 0.2f;

typedef __attribute__((ext_vector_type(2))) float v2f;
typedef __attribute__((ext_vector_type(8))) float v8f;

// ---------------------------------------------------------------------------
__global__ void k_init(float* __restrict__ ws) {
    int i = blockIdx.x * blockDim.x + threadIdx.x;
    if (i < 2176) ws[i] = 0.0f;
}

// ---------------------------------------------------------------------------
// Pass 1: mu_sum[n][c][k] = sum_p feat[n][c][p] * onehot[p][k]  via
// V_WMMA_F32_16X16X4_F32:  A = 16 channels x 4 pixels, B = onehot 4 px x 16 k.
// A layout: lanes 0-15 hold M=c (VGPR0=K0,VGPR1=K1), lanes 16-31 K2/K3.
// B layout: lanes 0-15 hold rows K0/K1 (N=lane), lanes 16-31 rows K2/K3.
// C layout: VGPR r -> (M = r + 8*(lane>=16), N = lane&15).
// ---------------------------------------------------------------------------
__global__ void __launch_bounds__(256) k_mu(const float* __restrict__ feat,
                                            const int* __restrict__ gt,
                                            float* __restrict__ ws) {
    __shared__ float smu[C_CH * K_CL];   // per-block partial mu_sum for this n

    const int n    = blockIdx.x / BLOCKS_PER_N;
    const int blk  = blockIdx.x % BLOCKS_PER_N;
    const int tid  = threadIdx.x;
    const int wave = tid >> 5;
    const int lane = tid & 31;
    const int l15  = lane & 15;          // channel-in-tile (A.M) and cluster (B.N)
    const int half = lane >> 4;          // 0: K-slots 0,1   1: K-slots 2,3

    for (int i = tid; i < C_CH * K_CL; i += 256) smu[i] = 0.0f;
    __syncthreads();

    const int PW = P_PIX / (BLOCKS_PER_N * 8);   // 1024 pixels per wave
    const int p0 = (blk * 8 + wave) * PW;

    const float* fA = feat + ((size_t)n * C_CH + l15) * P_PIX;        // channels 0..15
    const float* fB = feat + ((size_t)n * C_CH + l15 + 16) * P_PIX;   // channels 16..31
    const int*   g  = gt + (size_t)n * P_PIX;

    v8f acc0 = {};   // channels  0..15  x clusters
    v8f acc1 = {};   // channels 16..31  x clusters

    for (int p = p0; p < p0 + PW; p += 4) {
        const int pa = p + 2 * half;                 // pixel pair for this lane half
        v2f a0 = *(const v2f*)(fA + pa);             // b64, coalesced along p
        v2f a1 = *(const v2f*)(fB + pa);
        const int ka = g[pa];
        const int kb = g[pa + 1];
        v2f b;
        b.x = (ka == l15) ? 1.0f : 0.0f;             // onehot rows of B
        b.y = (kb == l15) ? 1.0f : 0.0f;
        acc0 = __builtin_amdgcn_wmma_f32_16x16x4_f32(false, a0, false, b,
                                                     (short)0, acc0, false, false);
        acc1 = __builtin_amdgcn_wmma_f32_16x16x4_f32(false, a1, false, b,
                                                     (short)0, acc1, false, false);
    }

    // Reduce 8 waves into LDS, then one set of global float atomics per block.
#pragma unroll
    for (int r = 0; r < 8; ++r) {
        const int c = r + 8 * half;
        unsafeAtomicAdd(&smu[c * K_CL + l15], acc0[r]);
        unsafeAtomicAdd(&smu[(c + 16) * K_CL + l15], acc1[r]);
    }
    __syncthreads();

    float* mu_sum = ws + (size_t)n * C_CH * K_CL;
    for (int i = tid; i < C_CH * K_CL; i += 256) unsafeAtomicAdd(&mu_sum[i], smu[i]);
}

// ---------------------------------------------------------------------------
// Pass 2: S1[n][k] = sum_{p: gt==k} ||f_p - mu_k||,  cnt[n][k] = #pixels.
// Each thread owns groups of 4 consecutive pixels -> b128 coalesced loads,
// mu for this image cached in LDS (512 floats).
// ---------------------------------------------------------------------------
__global__ void __launch_bounds__(256) k_var(const float* __restrict__ feat,
                                             const int* __restrict__ gt,
                                             const float* __restrict__ ws,
                                             float* __restrict__ wsS1,
                                             float* __restrict__ wsCnt) {
    __shared__ float smu[C_CH * K_CL];
    __shared__ float sS1[K_CL];
    __shared__ float sCnt[K_CL];

    const int n   = blockIdx.x / B2;
    const int blk = blockIdx.x % B2;
    const int tid = threadIdx.x;
    const float invP = 1.0f / (float)P_PIX;

    for (int i = tid; i < C_CH * K_CL; i += 256)
        smu[i] = ws[(size_t)n * C_CH * K_CL + i] * invP;
    if (tid < K_CL) { sS1[tid] = 0.0f; sCnt[tid] = 0.0f; }
    __syncthreads();

    const int CHUNK = P_PIX / B2;          // 2048 pixels per block
    const int pbase = blk * CHUNK;
    const float* f = feat + (size_t)n * C_CH * P_PIX;
    const int*   g = gt + (size_t)n * P_PIX;

    for (int p4 = pbase + tid * 4; p4 < pbase + CHUNK; p4 += 256 * 4) {
        const int4 kk = *(const int4*)(g + p4);
        float acc0 = 0.f, acc1 = 0.f, acc2 = 0.f, acc3 = 0.f;
#pragma unroll
        for (int c = 0; c < C_CH; ++c) {
            const float4 fv = *(const float4*)(f + (size_t)c * P_PIX + p4);
            const float* mrow = smu + c * K_CL;
            float d0 = fv.x - mrow[kk.x]; acc0 += d0 * d0;
            float d1 = fv.y - mrow[kk.y]; acc1 += d1 * d1;
            float d2 = fv.z - mrow[kk.z]; acc2 += d2 * d2;
            float d3 = fv.w - mrow[kk.w]; acc3 += d3 * d3;
        }
        unsafeAtomicAdd(&sS1[kk.x], sqrtf(acc0));
        unsafeAtomicAdd(&sS1[kk.y], sqrtf(acc1));
        unsafeAtomicAdd(&sS1[kk.z], sqrtf(acc2));
        unsafeAtomicAdd(&sS1[kk.w], sqrtf(acc3));
        unsafeAtomicAdd(&sCnt[kk.x], 1.0f);
        unsafeAtomicAdd(&sCnt[kk.y], 1.0f);
        unsafeAtomicAdd(&sCnt[kk.z], 1.0f);
        unsafeAtomicAdd(&sCnt[kk.w], 1.0f);
    }
    __syncthreads();

    if (tid < K_CL) {
        unsafeAtomicAdd(&wsS1[n * K_CL + tid], sS1[tid]);
        unsafeAtomicAdd(&wsCnt[n * K_CL + tid], sCnt[tid]);
    }
}

// ---------------------------------------------------------------------------
// Final: mu out, hinges, K x K distances, scalar losses. One block.
// ---------------------------------------------------------------------------
__global__ void __launch_bounds__(256) k_final(const float* __restrict__ ws,
                                               float* __restrict__ out) {
    __shared__ float smu[N_B * C_CH * K_CL];   // 2048
    __shared__ float s_sq[N_B * K_CL];         // ||mu||^2 per (n,k)
    __shared__ float rVar[64], rDist[64], rNorm[64];

    const int tid = threadIdx.x;
    const float invP = 1.0f / (float)P_PIX;

    for (int i = tid; i < N_B * C_CH * K_CL; i += 256) {
        const float m = ws[i] * invP;
        smu[i] = m;
        out[4 + i] = m;                         // mu, (n,c,k) row-major
    }
    __syncthreads();

    if (tid < N_B * K_CL) {
        const int n = tid >> 4, k = tid & 15;
        const float* mu_n = smu + n * C_CH * K_CL;
        float sq = 0.0f;
        for (int c = 0; c < C_CH; ++c) { float m = mu_n[c * K_CL + k]; sq += m * m; }
        s_sq[tid] = sq;
    }
    __syncthreads();

    if (tid < N_B * K_CL) {
        const int n = tid >> 4, k = tid & 15;
        const float* mu_n = smu + n * C_CH * K_CL;
        const float sqk = s_sq[tid];
        const float nrm = sqrtf(fmaxf(sqk, 0.0f));       // safe_sqrt

        // variance: v = (S1 + (P - cnt) * ||mu_k||) / P, hinge at delta_v
        const float S1  = ws[2048 + tid];
        const float cnt = ws[2112 + tid];
        const float v   = (S1 + ((float)P_PIX - cnt) * nrm) * invP;
        rVar[tid] = fmaxf(v - kDeltaV, 0.0f);

        // distance: mean over the K-1 other clusters, hinge at 2*delta_d
        float sumD = 0.0f;
        for (int j = 0; j < K_CL; ++j) {
            if (j == k) continue;
            float dot = 0.0f;
            for (int c = 0; c < C_CH; ++c)
                dot += mu_n[c * K_CL + k] * mu_n[c * K_CL + j];
            const float sqd = sqk + s_sq[(n << 4) | j] - 2.0f * dot;
            sumD += sqrtf(fmaxf(sqd, 0.0f));
        }
        rDist[tid] = fmaxf(2.0f * kDeltaD - sumD / (float)(K_CL - 1), 0.0f);
        rNorm[tid] = nrm;
    }
    __syncthreads();

    if (tid == 0) {
        float sv = 0.f, sd = 0.f, sn = 0.f;
        for (int i = 0; i < N_B * K_CL; ++i) { sv += rVar[i]; sd += rDist[i]; sn += rNorm[i]; }
        const float inv = 1.0f / (float)(N_B * K_CL);
        const float variance_loss      = sv * inv;
        const float distance_loss      = sd * inv;
        const float normalization_loss = sn * inv;
        out[0] = variance_loss + distance_loss + 0.001f * normalization_loss;
        out[1] = variance_loss;
        out[2] = distance_loss;
        out[3] = normalization_loss;
    }
}

// ---------------------------------------------------------------------------
extern "C" void kernel_launch(void* const* d_in, const int* in_sizes, int n_in,
                              void* d_out, int out_size, void* d_ws, size_t ws_size,
                              hipStream_t stream) {
    (void)in_sizes; (void)n_in; (void)out_size; (void)ws_size;
    const float* feat = (const float*)d_in[0];
    const int*   gt   = (const int*)d_in[1];
    float* out = (float*)d_out;
    float* ws  = (float*)d_ws;

    k_init<<<9, 256, 0, stream>>>(ws);
    k_mu<<<N_B * BLOCKS_PER_N, 256, 0, stream>>>(feat, gt, ws);
    k_var<<<N_B * B2, 256, 0, stream>>>(feat, gt, ws, ws + 2048, ws + 2112);
    k_final<<<1, 256, 0, stream>>>(ws, out);
}